// Attention_74938589380756
// MI455X (gfx1250) — compile-verified
//
#include <hip/hip_runtime.h>
#include <math.h>

#define B_    64
#define H_    32
#define KVH_  8
#define D_    128
#define BS_   16
#define MB_   128
#define G_    4
#define SCALE_ 0.08838834764831845f
#define NWAVES 8
#define PSTR  18   // padded LDS stride for P tile (distinct banks, even for b64)
#define RSTRIDE (KVH_ * D_)   // token-row stride in the paged caches (1024 floats)

typedef __attribute__((ext_vector_type(2))) float v2f;
typedef __attribute__((ext_vector_type(8))) float v8f;

__global__ __launch_bounds__(256)
void paged_attn_wmma_f32(const float* __restrict__ q,
                         const float* __restrict__ knew,
                         const float* __restrict__ vnew,
                         const float* __restrict__ kc,
                         const float* __restrict__ vc,
                         const int*   __restrict__ bt,
                         const int*   __restrict__ ctxl,
                         float*       __restrict__ out)
{
    __shared__ __align__(16) float qs[16 * D_];            // scaled Q, rows >= G zeroed (8 KB)
    __shared__ __align__(16) float ps[NWAVES][16 * PSTR];  // per-wave P scratch (transposed)
    __shared__ float wm[NWAVES * G_];
    __shared__ float wl[NWAVES * G_];
    __shared__ float wo[NWAVES * G_ * D_];                 // per-wave partial O (16 KB)
    __shared__ float snew[G_];                             // new-token scores

    const int bh   = blockIdx.x;
    const int b    = bh / KVH_;
    const int kvh  = bh % KVH_;
    const int tid  = threadIdx.x;
    const int wave = tid >> 5;
    const int lane = tid & 31;
    const int n    = lane & 15;   // tile column / A-row (token) index
    const int half = lane >> 4;
    const int kk   = half * 2;    // K-dim pair owned by this half-wave

    // ---- stage Q into LDS: qs[row][d], rows 0..3 = scaled q heads, rows 4..15 = 0 ----
    for (int i = tid; i < 16 * D_; i += 256) {
        const int r = i >> 7;
        const int d = i & 127;
        float val = 0.0f;
        if (r < G_) val = q[(size_t)b * (H_ * D_) + (size_t)(kvh * G_ + r) * D_ + d] * SCALE_;
        qs[i] = val;
    }
    __syncthreads();

    // wave-uniform scalars -> SGPRs so the chunk loop is a scalar branch (EXEC stays full)
    const int ctx     = __builtin_amdgcn_readfirstlane(ctxl[b]);
    const int nchunks = (ctx + 15) >> 4;   // cache positions 0..ctx-1 only

    const float* kvec = knew + ((size_t)b * KVH_ + kvh) * D_;
    const float* vvec = vnew + ((size_t)b * KVH_ + kvh) * D_;

    float m    = -INFINITY;
    float lsum = 0.0f;
    v8f oa[8];
    #pragma unroll
    for (int i = 0; i < 8; ++i) oa[i] = (v8f){0,0,0,0,0,0,0,0};

    const float* qrow = &qs[n * D_];

    // ---- flash-decode over pure cache chunks: wave handles chunks wave, wave+8, ... ----
    for (int c = wave; c < nchunks; c += NWAVES) {
        const int base = c << 4;
        const int blk  = __builtin_amdgcn_readfirstlane(bt[b * MB_ + c]);
        const float* tile = kc + (((size_t)blk * BS_) * KVH_ + kvh) * D_;   // K tile base
        const float* krow = tile + (size_t)n * RSTRIDE;                     // this lane's token row
        const float* vtile = vc + (((size_t)blk * BS_) * KVH_ + kvh) * D_;  // V tile base

        // prefetch next chunk of this wave (one tile ahead) while computing this one
        {
            const int cn = c + NWAVES;
            if (cn < nchunks) {
                const int blkn = __builtin_amdgcn_readfirstlane(bt[b * MB_ + cn]);
                const float* krn = kc + (((size_t)blkn * BS_ + n) * KVH_ + kvh) * D_ + half * 64;
                const float* vrn = vc + (((size_t)blkn * BS_ + n) * KVH_ + kvh) * D_ + half * 64;
                __builtin_prefetch(krn,      0, 3);
                __builtin_prefetch(krn + 32, 0, 3);
                __builtin_prefetch(vrn,      0, 3);
                __builtin_prefetch(vrn + 32, 0, 3);
            }
        }

        // ---- S^T[t][col] = sum_d K[t][d]*Qs[col][d] : batch loads, then WMMA chain ----
        v8f s = (v8f){0,0,0,0,0,0,0,0};
        #pragma unroll
        for (int hlf = 0; hlf < 2; ++hlf) {
            v2f ka[16], qb[16];
            #pragma unroll
            for (int i = 0; i < 16; ++i) {
                const int dc = (hlf * 16 + i) * 4;
                ka[i] = *(const v2f*)(krow + dc + kk);
                qb[i] = *(const v2f*)(qrow + dc + kk);
            }
            #pragma unroll
            for (int i = 0; i < 16; ++i)
                s = __builtin_amdgcn_wmma_f32_16x16x4_f32(
                        false, ka[i], false, qb[i], (short)0, s, false, false);
        }

        // ---- mask tokens >= ctx, per-column online softmax ----
        float tmax = -INFINITY;
        float pr[8];
        #pragma unroll
        for (int r = 0; r < 8; ++r) {
            const bool ok = (base + r + 8 * half) < ctx;
            const float sv = ok ? s[r] : -INFINITY;
            pr[r] = sv;
            tmax  = fmaxf(tmax, sv);
        }
        tmax = fmaxf(tmax, __shfl_xor(tmax, 16, 32));   // merge token halves per column

        const float mnew  = fmaxf(m, tmax);
        const float alpha = (mnew == m) ? 1.0f : __expf(m - mnew);

        float psum = 0.0f;
        #pragma unroll
        for (int r = 0; r < 8; ++r) {
            const float p_ = (pr[r] == -INFINITY) ? 0.0f : __expf(pr[r] - mnew);
            pr[r] = p_;
            psum += p_;
        }
        psum += __shfl_xor(psum, 16, 32);
        lsum = lsum * alpha + psum;
        m    = mnew;

        #pragma unroll
        for (int i = 0; i < 8; ++i) oa[i] *= alpha;

        // ---- re-shape P: C/D layout -> B layout via LDS (transposed: ps[col][token]) ----
        float* pw = &ps[wave][0];
        #pragma unroll
        for (int r = 0; r < 8; ++r)
            pw[n * PSTR + 8 * half + r] = pr[r];

        v2f bp[4];
        #pragma unroll
        for (int tci = 0; tci < 4; ++tci)
            bp[tci] = *(const v2f*)(pw + n * PSTR + tci * 4 + kk);

        // ---- O^T += V^T * P : batch V loads (32 dwords), then 16 WMMAs; twice ----
        #pragma unroll
        for (int pp = 0; pp < 2; ++pp) {
            v2f va[16];
            #pragma unroll
            for (int tt = 0; tt < 2; ++tt) {
                const int t0 = (pp * 2 + tt) * 4 + kk;
                const float* vr0 = vtile + (size_t)t0 * RSTRIDE;
                const float* vr1 = vr0 + RSTRIDE;
                #pragma unroll
                for (int dt = 0; dt < 8; ++dt) {
                    va[tt * 8 + dt].x = vr0[dt * 16 + n];
                    va[tt * 8 + dt].y = vr1[dt * 16 + n];
                }
            }
            #pragma unroll
            for (int tt = 0; tt < 2; ++tt) {
                const int tci = pp * 2 + tt;
                #pragma unroll
                for (int dt = 0; dt < 8; ++dt)
                    oa[dt] = __builtin_amdgcn_wmma_f32_16x16x4_f32(
                                 false, va[tt * 8 + dt], false, bp[tci], (short)0,
                                 oa[dt], false, false);
            }
        }
    }

    // ---- spill per-wave partial state ----
    if (half == 0 && n < G_) {
        wm[wave * G_ + n] = m;
        wl[wave * G_ + n] = lsum;
    }
    if (n < G_) {
        #pragma unroll
        for (int dt = 0; dt < 8; ++dt)
            #pragma unroll
            for (int r = 0; r < 8; ++r)
                wo[(wave * G_ + n) * D_ + dt * 16 + 8 * half + r] = oa[dt][r];
    }

    // ---- new-token scores s_new[g] = (q_g * SCALE) . k_new  (waves 0..3) ----
    if (tid < G_ * 32) {
        const int g   = tid >> 5;
        const int l32 = tid & 31;
        float part = 0.0f;
        #pragma unroll
        for (int d = 0; d < D_; d += 32) part += qs[g * D_ + d + l32] * kvec[d + l32];
        #pragma unroll
        for (int off = 16; off > 0; off >>= 1) part += __shfl_xor(part, off, 32);
        if (l32 == 0) snew[g] = part;
    }
    __syncthreads();

    // ---- merge 8 wave-partials + the new-token partial (m=snew, l=1, O=v_new) ----
    for (int idx = tid; idx < G_ * D_; idx += 256) {
        const int g = idx >> 7;
        const int d = idx & 127;
        float M = snew[g];
        #pragma unroll
        for (int w = 0; w < NWAVES; ++w) M = fmaxf(M, wm[w * G_ + g]);
        const float sc9 = __expf(snew[g] - M);
        float L   = sc9;
        float acc = vvec[d] * sc9;
        #pragma unroll
        for (int w = 0; w < NWAVES; ++w) {
            const float mw = wm[w * G_ + g];
            const float sc = (mw == -INFINITY) ? 0.0f : __expf(mw - M);
            L   += wl[w * G_ + g] * sc;
            acc += wo[(w * G_ + g) * D_ + d] * sc;
        }
        out[(size_t)b * (H_ * D_) + (size_t)(kvh * G_ + g) * D_ + d] = acc / L;
    }
}

extern "C" void kernel_launch(void* const* d_in, const int* in_sizes, int n_in,
                              void* d_out, int out_size, void* d_ws, size_t ws_size,
                              hipStream_t stream) {
    const float* q    = (const float*)d_in[0];
    const float* k    = (const float*)d_in[1];
    const float* v    = (const float*)d_in[2];
    const float* kc   = (const float*)d_in[3];
    const float* vc   = (const float*)d_in[4];
    const int*   bt   = (const int*)d_in[5];
    const int*   ctxl = (const int*)d_in[6];
    float*       out  = (float*)d_out;

    dim3 grid(B_ * KVH_);   // one workgroup per (batch, kv-head)
    dim3 block(256);        // 8 waves (wave32)
    paged_attn_wmma_f32<<<grid, block, 0, stream>>>(q, k, v, kc, vc, bt, ctxl, out);
}